// RoIAlign_35519379537988
// MI455X (gfx1250) — compile-verified
//
#include <hip/hip_runtime.h>

typedef float v2f __attribute__((ext_vector_type(2)));
typedef float v8f __attribute__((ext_vector_type(8)));

#define AH 7
#define AW 7
#define SCALE 0.125f

// One wave (32 threads) per block. Block = (roi n, 16-channel group).
// Lanes 0..15 take the (wi) column taps, lanes 16..31 the (wi+1) column taps,
// each lane pre-multiplies its two taps by their bilinear weights, and
// V_WMMA_F32_16X16X4_F32 with an all-ones A performs the 4-term cross-lane
// sum per channel column. Result staged in LDS, written back coalesced b128.
__global__ __launch_bounds__(32) void roialign_wmma_kernel(
    const float* __restrict__ feats,   // [B, C, H, W]
    const float* __restrict__ rois,    // [N, 7] (b, cls, x1, y1, x2, y2, score)
    float* __restrict__ out,           // [N, C, 7, 7]
    int C, int H, int W)
{
    __shared__ __align__(16) float tile[16 * AH * AW];   // 16 channels x 49 points

    const int lane   = threadIdx.x;          // 0..31
    const int groups = C >> 4;               // channel groups of 16
    const int n      = blockIdx.x / groups;
    const int c0     = (blockIdx.x % groups) << 4;

    const float* r = rois + (size_t)n * 7;
    const int   b  = (int)r[0];
    const float x1 = r[2] * SCALE;
    const float y1 = r[3] * SCALE;
    const float x2 = r[4] * SCALE;
    const float y2 = r[5] * SCALE;
    const float roi_w = fmaxf(x2 - x1, 0.0f);
    const float roi_h = fmaxf(y2 - y1, 0.0f);
    const float bin_h = roi_h / (float)(AH - 1);
    const float bin_w = roi_w / (float)(AW - 1);

    const int  cl      = lane & 15;          // local channel 0..15
    const bool hi_half = (lane >= 16);       // upper half takes wi+1 column
    const float* plane = feats + ((size_t)(b * C + c0 + cl)) * (size_t)(H * W);

    const v2f a  = {1.0f, 1.0f};             // all-ones A: WMMA = per-column 4-term sum
    const v8f cz = {};

    for (int ph = 0; ph < AH; ++ph) {
        const float h      = y1 + (float)ph * bin_h;
        const float hstart = fminf(floorf(h), (float)(H - 2));
        const float hr     = h - hstart;
        const int   hi     = (int)fminf(fmaxf(hstart, 0.0f), (float)(H - 2));
        const bool  hok    = (h >= 0.0f) && (h < (float)H);

        for (int pw = 0; pw < AW; ++pw) {
            const float w      = x1 + (float)pw * bin_w;
            const float wstart = fminf(floorf(w), (float)(W - 2));
            const float wr     = w - wstart;
            const int   wi     = (int)fminf(fmaxf(wstart, 0.0f), (float)(W - 2));
            const bool  ok     = hok && (w >= 0.0f) && (w < (float)W);
            const float m      = ok ? 1.0f : 0.0f;

            // this lane's two bilinear taps (top and bottom row of its column)
            const int   col = wi + (hi_half ? 1 : 0);
            const float vt  = plane[hi * W + col];
            const float vb  = plane[(hi + 1) * W + col];
            const float wrc = hi_half ? wr : (1.0f - wr);
            const float u0  = vt * ((1.0f - hr) * wrc) * m;   // (hi  , col) contribution
            const float u1  = vb * (hr * wrc) * m;            // (hi+1, col) contribution

            v2f bm = {u0, u1};
            // D[m, nCol] = sum over the 4 slots of column nCol (order-independent)
            v8f d = __builtin_amdgcn_wmma_f32_16x16x4_f32(
                false, a, false, bm, (short)0, cz, false, false);

            if (!hi_half)
                tile[cl * (AH * AW) + ph * AW + pw] = d[0];
        }
    }
    __syncthreads();

    // Coalesced b128 writeback: (n, c0..c0+15, :, :) is 784 contiguous floats.
    float4*       dst = (float4*)(out + ((size_t)n * C + c0) * (AH * AW));
    const float4* src = (const float4*)tile;
    for (int i = lane; i < (16 * AH * AW) / 4; i += 32)
        dst[i] = src[i];
}

extern "C" void kernel_launch(void* const* d_in, const int* in_sizes, int n_in,
                              void* d_out, int out_size, void* d_ws, size_t ws_size,
                              hipStream_t stream) {
    const float* feats = (const float*)d_in[0];
    const float* rois  = (const float*)d_in[1];
    float*       out   = (float*)d_out;

    const int C = 512, H = 60, W = 80;
    const int N = in_sizes[1] / 7;            // 1024 RoIs

    dim3 grid((unsigned)(N * (C / 16)));      // 32768 single-wave blocks
    dim3 block(32);
    hipLaunchKernelGGL(roialign_wmma_kernel, grid, block, 0, stream,
                       feats, rois, out, C, H, W);
}